// STABlock_88270167867972
// MI455X (gfx1250) — compile-verified
//
#include <hip/hip_runtime.h>
#include <math.h>

// ---------------------------------------------------------------------------
// STABlock for MI455X (gfx1250, wave32). Flash-style spatial attention with
// V_WMMA_F32_16X16X4_F32 tiles (fp32-exact matrix path) and *deferred*
// softmax normalization (shift-invariant, scores are O(1) so no max needed):
// the inner loop is 8 WMMA + 16 LDS ops + 8 exp per 16x16 key tile, with a
// single cross-lane row-sum reduction after the loop.
// ---------------------------------------------------------------------------

typedef __attribute__((ext_vector_type(2))) float v2f;
typedef __attribute__((ext_vector_type(8))) float v8f;

#define WMMA_F32(A, B, C) \
  __builtin_amdgcn_wmma_f32_16x16x4_f32(false, (A), false, (B), (short)0, (C), false, false)

static constexpr int kB    = 2;
static constexpr int kC    = 64;
static constexpr int kCI   = 32;
static constexpr int kS3   = 3136;    // T*H*W
static constexpr int kN    = 6272;    // CI*H*W  ("N" of the raw reshape)
static constexpr int kFlat = 100352;  // CI*T*H*W per batch
static constexpr int kNT   = 392;     // kN / 16 key tiles

// ---------------------------------------------------------------------------
// Kernel 1: all six 1x1x1-conv projections fused. One thread per (b,o,s);
// x loads are coalesced over s, each x element is read once and feeds 6 FMAs.
// Outputs stored in raw (CI,T,H,W) flat order (the order the reference's
// .view() interprets).
// ---------------------------------------------------------------------------
__global__ __launch_bounds__(256) void proj6_kernel(
    const float* __restrict__ x,
    const float* __restrict__ W0, const float* __restrict__ B0,
    const float* __restrict__ W1, const float* __restrict__ B1,
    const float* __restrict__ W2, const float* __restrict__ B2,
    const float* __restrict__ W3, const float* __restrict__ B3,
    const float* __restrict__ W4, const float* __restrict__ B4,
    const float* __restrict__ W5, const float* __restrict__ B5,
    float* __restrict__ o0, float* __restrict__ o1, float* __restrict__ o2,
    float* __restrict__ o3, float* __restrict__ o4, float* __restrict__ o5)
{
  const int idx = blockIdx.x * 256 + threadIdx.x;   // == b*kFlat + o*kS3 + s
  const int s = idx % kS3;
  const int o = (idx / kS3) % kCI;
  const int b = idx / kFlat;
  const float* xb = x + b * (kC * kS3) + s;
  const float* w0 = W0 + o * kC; const float* w1 = W1 + o * kC;
  const float* w2 = W2 + o * kC; const float* w3 = W3 + o * kC;
  const float* w4 = W4 + o * kC; const float* w5 = W5 + o * kC;
  float a0 = B0[o], a1 = B1[o], a2 = B2[o], a3 = B3[o], a4 = B4[o], a5 = B5[o];
#pragma unroll 8
  for (int c = 0; c < kC; ++c) {
    const float xv = xb[c * kS3];
    a0 = fmaf(w0[c], xv, a0);
    a1 = fmaf(w1[c], xv, a1);
    a2 = fmaf(w2[c], xv, a2);
    a3 = fmaf(w3[c], xv, a3);
    a4 = fmaf(w4[c], xv, a4);
    a5 = fmaf(w5[c], xv, a5);
  }
  o0[idx] = a0; o1[idx] = a1; o2[idx] = a2;
  o3[idx] = a3; o4[idx] = a4; o5[idx] = a5;
}

// ---------------------------------------------------------------------------
// Kernel 2: spatial branch, streaming attention. One wave per 16-query block
// (B*392 = 784 waves). Scores S = Q K^T via 4x WMMA_F32_16x16x4; p = exp(S/4)
// with deferred normalization (per-lane partial row sums, reduced once after
// the loop); P relayout C-layout -> A-layout through a per-wave LDS tile;
// O accumulated with 4x WMMA into the same v8f accumulator.
//   q/k flats:  Q[n,t] = q[t*kN + n]          (raw reshape (T, N))
//   v flat:     V[m,t] = v[t*kN + m]
//   y output:   ys[n,t] -> y[t*kN + n]  (== raw (CI,T,H,W) flat)
// ---------------------------------------------------------------------------
__global__ __launch_bounds__(128) void spatial_attn_kernel(
    const float* __restrict__ qf, const float* __restrict__ kf,
    const float* __restrict__ vf, float* __restrict__ y)
{
  __shared__ float Pt[4][16][17];     // per-wave exp(score) tile, padded
  const int lane = threadIdx.x & 31;
  const int wv   = threadIdx.x >> 5;
  const int gw   = blockIdx.x * 4 + wv;
  const int b    = gw / kNT;
  const int n0   = (gw - b * kNT) * 16;
  const float* q = qf + b * kFlat;
  const float* k = kf + b * kFlat;
  const float* v = vf + b * kFlat;
  float* yb      = y + b * kFlat;
  const int col  = lane & 15;
  const int half = lane >> 4;

  // Q fragment for the whole block: A-layout, a[r] holds K = 4*s4 + 2*half + r
  float qa[8];
#pragma unroll
  for (int s4 = 0; s4 < 4; ++s4)
#pragma unroll
    for (int r = 0; r < 2; ++r)
      qa[s4 * 2 + r] = q[(s4 * 4 + half * 2 + r) * kN + n0 + col];

  v8f acc = {0.f, 0.f, 0.f, 0.f, 0.f, 0.f, 0.f, 0.f};
  float psum[8];
#pragma unroll
  for (int r = 0; r < 8; ++r) psum[r] = 0.f;

  for (int mt = 0; mt < kNT; ++mt) {
    const int m0 = mt * 16;
    float kb[8], vb[8];
#pragma unroll
    for (int s4 = 0; s4 < 4; ++s4)
#pragma unroll
      for (int r = 0; r < 2; ++r) {
        const int kk = s4 * 4 + half * 2 + r;
        kb[s4 * 2 + r] = k[kk * kN + m0 + col];          // B1[kk,m] = K^T
        vb[s4 * 2 + r] = v[col * kN + m0 + kk];          // B2[kk,t] = V
      }
    // S tile = Q * K^T  (C-layout: row = r + 8*half (query), col = key)
    v8f st = {0.f, 0.f, 0.f, 0.f, 0.f, 0.f, 0.f, 0.f};
#pragma unroll
    for (int s4 = 0; s4 < 4; ++s4) {
      v2f a;  a.x  = qa[s4 * 2]; a.y  = qa[s4 * 2 + 1];
      v2f bb; bb.x = kb[s4 * 2]; bb.y = kb[s4 * 2 + 1];
      st = WMMA_F32(a, bb, st);
    }
    // p = exp(scale * s); accumulate per-lane partial row sums (normalization
    // deferred: softmax is shift-invariant and scores are O(1) in fp32 range)
#pragma unroll
    for (int r = 0; r < 8; ++r) {
      const float p = __expf(st[r] * 0.25f);             // scale = 1/sqrt(T)
      psum[r] += p;
      Pt[wv][half * 8 + r][col] = p;                     // C-layout -> LDS
    }
    // read P back in A-layout (same-wave LDS, in-order, no barrier needed)
#pragma unroll
    for (int s4 = 0; s4 < 4; ++s4) {
      v2f a;
      a.x = Pt[wv][col][s4 * 4 + half * 2 + 0];
      a.y = Pt[wv][col][s4 * 4 + half * 2 + 1];
      v2f bb; bb.x = vb[s4 * 2]; bb.y = vb[s4 * 2 + 1];
      acc = WMMA_F32(a, bb, acc);                        // O += P * V
    }
  }
  // one cross-lane reduction for the row sums, then normalize and store
#pragma unroll
  for (int r = 0; r < 8; ++r) {
    float ts = psum[r];
    ts += __shfl_xor(ts, 1, 32);
    ts += __shfl_xor(ts, 2, 32);
    ts += __shfl_xor(ts, 4, 32);
    ts += __shfl_xor(ts, 8, 32);
    yb[col * kN + n0 + half * 8 + r] = acc[r] / ts;
  }
}

// ---------------------------------------------------------------------------
// Kernel 3: temporal branch. One block per batch, 8 waves.
// Phase 1: G[t,s] = sum_n q[n*16+t]*k[n*16+s] via WMMA (1568 K=4 chunks split
//          across waves), LDS reduction + exact softmax in wave 0.
// Phase 2: yt[t,n] = attn * V over 392 n-tiles; "+=" into y (after spatial).
// ---------------------------------------------------------------------------
__global__ __launch_bounds__(256) void temporal_attn_kernel(
    const float* __restrict__ qf, const float* __restrict__ kf,
    const float* __restrict__ vf, float* __restrict__ y)
{
  __shared__ float Gp[8][16][17];
  __shared__ float At[16][17];
  const int lane = threadIdx.x & 31;
  const int wv   = threadIdx.x >> 5;
  const int b    = blockIdx.x;
  const float* q = qf + b * kFlat;
  const float* k = kf + b * kFlat;
  const float* v = vf + b * kFlat;
  float* yb      = y + b * kFlat;
  const int col  = lane & 15;
  const int half = lane >> 4;

  // partial Gram matrix: inner dim n, 1568 chunks of 4, strided over 8 waves
  v8f g = {0.f, 0.f, 0.f, 0.f, 0.f, 0.f, 0.f, 0.f};
  for (int nc = wv; nc < 1568; nc += 8) {
    const int base = nc * 64;                     // (nc*4 + kk)*16 + col
    v2f a, bb;
    a.x  = q[base + (half * 2 + 0) * 16 + col];
    a.y  = q[base + (half * 2 + 1) * 16 + col];
    bb.x = k[base + (half * 2 + 0) * 16 + col];
    bb.y = k[base + (half * 2 + 1) * 16 + col];
    g = WMMA_F32(a, bb, g);
  }
#pragma unroll
  for (int r = 0; r < 8; ++r) Gp[wv][half * 8 + r][col] = g[r];
  __syncthreads();

  if (threadIdx.x < 32) {   // wave 0: reduce partials + exact softmax
#pragma unroll
    for (int r = 0; r < 8; ++r) {
      float sc = 0.f;
#pragma unroll
      for (int j = 0; j < 8; ++j) sc += Gp[j][half * 8 + r][col];
      sc *= 0.25f;
      float mx = sc;
      mx = fmaxf(mx, __shfl_xor(mx, 1, 32));
      mx = fmaxf(mx, __shfl_xor(mx, 2, 32));
      mx = fmaxf(mx, __shfl_xor(mx, 4, 32));
      mx = fmaxf(mx, __shfl_xor(mx, 8, 32));
      const float p = __expf(sc - mx);
      float ts = p;
      ts += __shfl_xor(ts, 1, 32);
      ts += __shfl_xor(ts, 2, 32);
      ts += __shfl_xor(ts, 4, 32);
      ts += __shfl_xor(ts, 8, 32);
      At[half * 8 + r][col] = p / ts;
    }
  }
  __syncthreads();

  // attn fragment in A-layout (constant over all n-tiles)
  float pa[8];
#pragma unroll
  for (int s4 = 0; s4 < 4; ++s4)
#pragma unroll
    for (int r = 0; r < 2; ++r)
      pa[s4 * 2 + r] = At[col][s4 * 4 + half * 2 + r];

  for (int nt = wv; nt < kNT; nt += 8) {
    const int n0 = nt * 16;
    v8f o = {0.f, 0.f, 0.f, 0.f, 0.f, 0.f, 0.f, 0.f};
#pragma unroll
    for (int s4 = 0; s4 < 4; ++s4) {
      v2f a;  a.x = pa[s4 * 2]; a.y = pa[s4 * 2 + 1];
      v2f bb;
      bb.x = v[(n0 + col) * 16 + s4 * 4 + half * 2 + 0];
      bb.y = v[(n0 + col) * 16 + s4 * 4 + half * 2 + 1];
      o = WMMA_F32(a, bb, o);
    }
#pragma unroll
    for (int r = 0; r < 8; ++r) {
      const int fi = (n0 + col) * 16 + half * 8 + r;    // yt[t,n] -> n*16+t
      yb[fi] += o[r];
    }
  }
}

// ---------------------------------------------------------------------------
// Kernel 4: output projection (64x32) + bias + residual.
// ---------------------------------------------------------------------------
__global__ __launch_bounds__(256) void out_proj_kernel(
    const float* __restrict__ x, const float* __restrict__ Ww,
    const float* __restrict__ bw, const float* __restrict__ y,
    float* __restrict__ out)
{
  const int idx = blockIdx.x * 256 + threadIdx.x;  // b*kC*kS3 + o*kS3 + s
  const int s = idx % kS3;
  const int o = (idx / kS3) % kC;
  const int b = idx / (kC * kS3);
  const float* yb = y + b * kFlat + s;
  const float* wr = Ww + o * kCI;
  float a = bw[o] + x[idx];
#pragma unroll 8
  for (int ci = 0; ci < kCI; ++ci)
    a = fmaf(wr[ci], yb[ci * kS3], a);
  out[idx] = a;
}

// ---------------------------------------------------------------------------
extern "C" void kernel_launch(void* const* d_in, const int* in_sizes, int n_in,
                              void* d_out, int out_size, void* d_ws, size_t ws_size,
                              hipStream_t stream) {
  const float* x   = (const float*)d_in[0];
  const float* Wqs = (const float*)d_in[1];  const float* bqs = (const float*)d_in[2];
  const float* Wks = (const float*)d_in[3];  const float* bks = (const float*)d_in[4];
  const float* Wvs = (const float*)d_in[5];  const float* bvs = (const float*)d_in[6];
  const float* Wqt = (const float*)d_in[7];  const float* bqt = (const float*)d_in[8];
  const float* Wkt = (const float*)d_in[9];  const float* bkt = (const float*)d_in[10];
  const float* Wvt = (const float*)d_in[11]; const float* bvt = (const float*)d_in[12];
  const float* Ww  = (const float*)d_in[13]; const float* bw  = (const float*)d_in[14];

  float* ws  = (float*)d_ws;                 // 1,404,928 floats (~5.6 MB)
  float* Pqs = ws + 0 * kB * kFlat;
  float* Pks = ws + 1 * kB * kFlat;
  float* Pvs = ws + 2 * kB * kFlat;
  float* Pqt = ws + 3 * kB * kFlat;
  float* Pkt = ws + 4 * kB * kFlat;
  float* Pvt = ws + 5 * kB * kFlat;
  float* Y   = ws + 6 * kB * kFlat;

  // 1) six fused projections (B*CI*S3 = 200,704 threads)
  proj6_kernel<<<(kB * kCI * kS3) / 256, 256, 0, stream>>>(
      x, Wqs, bqs, Wks, bks, Wvs, bvs, Wqt, bqt, Wkt, bkt, Wvt, bvt,
      Pqs, Pks, Pvs, Pqt, Pkt, Pvt);

  // 2) spatial streaming attention: B*392 query-block waves, 4 waves/block
  spatial_attn_kernel<<<(kB * kNT) / 4, 128, 0, stream>>>(Pqs, Pks, Pvs, Y);

  // 3) temporal attention, accumulates into Y (stream-ordered after spatial)
  temporal_attn_kernel<<<kB, 256, 0, stream>>>(Pqt, Pkt, Pvt, Y);

  // 4) output projection + residual (B*C*S3 = 401,408 threads)
  out_proj_kernel<<<(kB * kC * kS3) / 256, 256, 0, stream>>>(x, Ww, bw, Y, (float*)d_out);
}